// Gaussian_63642825392442
// MI455X (gfx1250) — compile-verified
//
#include <hip/hip_runtime.h>

#define DORIG 100
#define DP    112          // padded to 7 tiles of 16
#define LDA   113          // LDS row stride (floats) to stagger banks
#define NT    7
#define LOG2PI 1.8378770664093453f

typedef __attribute__((ext_vector_type(2))) float v2f;
typedef __attribute__((ext_vector_type(8))) float v8f;

// ---------------- Kernel 1: cov = aux @ aux^T, padded 112x112 with identity ----
__global__ void build_cov_kernel(const float* __restrict__ logd,
                                 const float* __restrict__ lt,
                                 float* __restrict__ covp) {
    int idx = blockIdx.x * blockDim.x + threadIdx.x;
    if (idx >= DP * DP) return;
    int i = idx / DP;
    int j = idx - i * DP;
    float v;
    if (i < DORIG && j < DORIG) {
        int lo = (i < j) ? i : j;
        int bi = i * (i - 1) / 2;
        int bj = j * (j - 1) / 2;
        float di = expf(logd[i]);
        float dj = expf(logd[j]);
        float s = 0.0f;
        for (int k = 0; k <= lo; ++k) {
            float ai = (k < i) ? lt[bi + k] : di;  // k<=lo<=i
            float aj = (k < j) ? lt[bj + k] : dj;
            s = fmaf(ai, aj, s);
        }
        v = s;
    } else {
        v = (i == j) ? 1.0f : 0.0f;   // identity padding
    }
    covp[idx] = v;
}

// ---------------- Kernel 2: per-sample masked Cholesky + solve + NLL ----------
__launch_bounds__(128)
__global__ void chol_nll_kernel(const float* __restrict__ x,
                                const int*   __restrict__ mask,
                                const float* __restrict__ mu,
                                const float* __restrict__ covp,
                                float* __restrict__ out) {
    __shared__ float A[DP * LDA];   // ~50.6 KB
    __shared__ float rs[DP];
    __shared__ float mv[DP];
    __shared__ float red[32];

    const int b    = blockIdx.x;
    const int tid  = threadIdx.x;
    const int lane = tid & 31;
    // force wave id into an SGPR -> scalar branches, statically full EXEC at WMMA
    const int wave = __builtin_amdgcn_readfirstlane(tid >> 5);

    // ---- masks + residual ----
    if (tid < DP) {
        float m = (tid < DORIG) ? (float)mask[b * DORIG + tid] : 0.0f;
        mv[tid] = m;
        rs[tid] = (tid < DORIG) ? (x[b * DORIG + tid] - mu[tid]) * m : 0.0f;
    }
    __syncthreads();

    // ---- masked covariance into LDS: cov*mi*mj + I*(1-mi) (padding -> identity)
    for (int i = 0; i < DP; ++i) {
        float mi = mv[i];
        for (int j = tid; j < DP; j += 128) {      // at most one j per thread (112<128)
            float v = covp[i * DP + j] * mi * mv[j];
            if (i == j) v += 1.0f - mi;
            A[i * LDA + j] = v;
        }
    }
    __syncthreads();

    // ---- blocked right-looking Cholesky, NB=16 ----
    for (int kb = 0; kb < NT; ++kb) {
        const int o = kb * 16;

        // 1) factor diagonal 16x16 block: wave 0, lanes in lockstep
        if (wave == 0) {
            for (int j = 0; j < 16; ++j) {
                __builtin_amdgcn_wave_barrier();
                float d   = sqrtf(A[(o + j) * LDA + (o + j)]);
                float inv = 1.0f / d;
                if (lane == j) A[(o + j) * LDA + (o + j)] = d;
                __builtin_amdgcn_wave_barrier();
                if (lane > j && lane < 16) {
                    int   r   = o + lane;
                    float lij = A[r * LDA + (o + j)] * inv;
                    A[r * LDA + (o + j)] = lij;        // scaled column (one ds_store, all lanes)
                    __builtin_amdgcn_wave_barrier();
                    for (int k = j + 1; k <= lane; ++k)  // rank-1 trailing (lower only)
                        A[r * LDA + (o + k)] -= lij * A[(o + k) * LDA + (o + j)];
                }
            }
        }
        __syncthreads();

        // 2) panel solve: L21 = A21 * L11^-T ; one row per thread (<=96 rows)
        const int rowsBelow = (NT - 1 - kb) * 16;
        if (tid < rowsBelow) {
            const int r = o + 16 + tid;
            float row[16];
            #pragma unroll
            for (int j = 0; j < 16; ++j) row[j] = A[r * LDA + o + j];
            for (int j = 0; j < 16; ++j) {
                float v = row[j];
                for (int k = 0; k < j; ++k)
                    v -= row[k] * A[(o + j) * LDA + (o + k)];
                v *= 1.0f / A[(o + j) * LDA + (o + j)];
                row[j] = v;
            }
            #pragma unroll
            for (int j = 0; j < 16; ++j) A[r * LDA + o + j] = row[j];
        }
        __syncthreads();

        // 3) trailing SYRK update via WMMA: A[I][J] -= L[I][kb] * L[J][kb]^T
        //    tile pairs round-robined over the 4 waves; guard is scalar (wave in SGPR)
        {
            int p = 0;
            for (int J = kb + 1; J < NT; ++J) {
                for (int I = J; I < NT; ++I) {
                    if ((p & 3) == wave) {
                        const int m    = lane & 15;
                        const int half = lane >> 4;          // 0 or 1
                        const int kofs = half * 2;           // A/B frag K placement
                        const int colC = J * 16 + m;
                        v8f c;
                        #pragma unroll
                        for (int vv = 0; vv < 8; ++vv) {     // C/D: row = vv (+8 upper half)
                            int rr = I * 16 + vv + (half << 3);
                            c[vv] = A[rr * LDA + colC];
                        }
                        #pragma unroll
                        for (int kc = 0; kc < 4; ++kc) {     // K=16 as 4 chained K=4 WMMAs
                            int col0 = o + kc * 4 + kofs;
                            v2f a, bb;
                            a.x  = -A[(I * 16 + m) * LDA + col0];      // -L_I (f32 WMMA: C-neg only)
                            a.y  = -A[(I * 16 + m) * LDA + col0 + 1];
                            bb.x =  A[(J * 16 + m) * LDA + col0];      // L_J^T chunk
                            bb.y =  A[(J * 16 + m) * LDA + col0 + 1];
                            c = __builtin_amdgcn_wmma_f32_16x16x4_f32(
                                    false, a, false, bb, (short)0, c, false, false);
                        }
                        #pragma unroll
                        for (int vv = 0; vv < 8; ++vv) {
                            int rr = I * 16 + vv + (half << 3);
                            A[rr * LDA + colC] = c[vv];
                        }
                    }
                    ++p;
                }
            }
        }
        __syncthreads();
    }

    // ---- forward substitution + NLL assembly: wave 0 only, wave-synchronous ----
    // Each lane owns rows {lane, lane+32, lane+64, lane+96}; DS ops are in-order
    // per wave so no workgroup barriers are needed (waves 1-3 are done).
    if (wave == 0) {
        float quad = 0.0f;
        for (int j = 0; j < DORIG; ++j) {
            __builtin_amdgcn_wave_barrier();
            float s = rs[j] * (1.0f / A[j * LDA + j]);   // uniform broadcast read
            if (lane == 0) quad += s * s;
            #pragma unroll
            for (int q = 0; q < 4; ++q) {
                int i = lane + q * 32;
                if (i > j && i < DP)
                    rs[i] -= A[i * LDA + j] * s;
            }
            __builtin_amdgcn_wave_barrier();
        }

        // logdet + observed count, per-lane partials
        float ld2 = 0.0f, kk = 0.0f;
        for (int i = lane; i < DORIG; i += 32) {
            ld2 += logf(A[i * LDA + i]);     // masked dims: log(1)=0
            kk  += mv[i];
        }
        red[lane] = 2.0f * ld2 + kk * LOG2PI + ((lane == 0) ? quad : 0.0f);
        __builtin_amdgcn_wave_barrier();
        if (lane == 0) {
            float s = 0.0f;
            for (int t = 0; t < 32; ++t) s += red[t];
            out[b] = 0.5f * s;
        }
    }
}

extern "C" void kernel_launch(void* const* d_in, const int* in_sizes, int n_in,
                              void* d_out, int out_size, void* d_ws, size_t ws_size,
                              hipStream_t stream) {
    const float* x    = (const float*)d_in[0];
    const int*   mask = (const int*)  d_in[1];
    const float* mu   = (const float*)d_in[2];
    const float* logd = (const float*)d_in[3];
    const float* lt   = (const float*)d_in[4];
    float* out  = (float*)d_out;
    float* covp = (float*)d_ws;              // 112*112*4 = 50176 B scratch

    const int Bn = in_sizes[0] / DORIG;

    build_cov_kernel<<<(DP * DP + 255) / 256, 256, 0, stream>>>(logd, lt, covp);
    chol_nll_kernel<<<Bn, 128, 0, stream>>>(x, mask, mu, covp, out);
}